// HardNegativeInfoNCELoss_86414741995691
// MI455X (gfx1250) — compile-verified
//
#include <hip/hip_runtime.h>

// ---------------------------------------------------------------------------
// HardNegativeInfoNCELoss for MI455X (gfx1250, wave32)
// B=64, Q=32, S=256, D=128, M=5, TAU=0.02
// Per-WG: d[c] staged once (async global->LDS), 4 query batches b computed
// against it via V_WMMA_F32_16X16X4_F32 + fused segmented modality max.
// ---------------------------------------------------------------------------

#define TAU_F    0.02f
#define INV_TAU  50.0f
#define QROWS    32
#define SROWS    256
#define DDIM     128
#define MMOD     5
#define BATCH    64
#define B_PER_WG 4
#define LDSTRIDE 132   // 128 + 4 pad: row stride advances 4 LDS banks per row

typedef __attribute__((ext_vector_type(2))) float v2f;
typedef __attribute__((ext_vector_type(8))) float v8f;

// Order-preserving float -> u32 map so we can use ds_max_u32 atomics in LDS.
__device__ __forceinline__ unsigned encOrd(float f) {
  unsigned u = __float_as_uint(f);
  return (u & 0x80000000u) ? ~u : (u | 0x80000000u);
}
__device__ __forceinline__ float decOrd(unsigned u) {
  return (u & 0x80000000u) ? __uint_as_float(u ^ 0x80000000u)
                           : __uint_as_float(~u);
}

// Async copy of one 16-byte chunk from global to LDS (no VGPR round-trip).
// LDS byte offset = low 32 bits of the generic shared-memory address
// (flat LDS aperture keeps the offset in addr[31:0] per the CDNA5 ISA).
__device__ __forceinline__ void asyncCopy16(void* lds_ptr, const void* gptr) {
  unsigned lds = (unsigned)(uintptr_t)lds_ptr;
  unsigned long long ga = (unsigned long long)(uintptr_t)gptr;
  asm volatile("global_load_async_to_lds_b128 %0, %1, off"
               :: "v"(lds), "v"(ga)
               : "memory");
}
__device__ __forceinline__ void waitAsync0() {
  asm volatile("s_wait_asynccnt 0x0" ::: "memory");
}

__global__ __launch_bounds__(256) void score_kernel(
    const float* __restrict__ qemb,           // (B, Q, D)
    const float* __restrict__ demb,           // (B, S, D)
    const int* __restrict__ modal,            // (B, S)
    const unsigned char* __restrict__ qmask,  // (B, Q) bool
    float* __restrict__ simg,                 // out (B, C)
    float* __restrict__ simsg) {              // out (B, C, M)
  extern __shared__ float smem[];
  float*    qs  = smem;                                 // 32*132 floats
  float*    dsm = smem + QROWS * LDSTRIDE;              // 256*132 floats
  unsigned* mx  = (unsigned*)(dsm + SROWS * LDSTRIDE);  // 32*5 u32
  float*    red = (float*)(mx + QROWS * MMOD);          // 32*(5+2) floats

  const int c   = blockIdx.x;
  const int b0  = blockIdx.y * B_PER_WG;
  const int tid = threadIdx.x;
  const int w   = tid >> 5;   // wave 0..7
  const int l   = tid & 31;   // lane
  const int r   = l & 15;     // row/col within 16-wide tile
  const int h   = l >> 4;     // lane half

  // ---- stage d[c] (256x128) into padded LDS once, asynchronously ----
  const float4* dg = (const float4*)(demb + (size_t)c * SROWS * DDIM);
  for (int i = tid; i < SROWS * DDIM / 4; i += 256) {
    int row = i >> 5;              // 32 float4 per 128-float row
    int col = (i & 31) << 2;
    asyncCopy16(&dsm[row * LDSTRIDE + col], dg + i);
  }

  // loop-invariant per-lane state
  const int sA = 32 * w + r;
  const int sB = sA + 16;
  const int mA = modal[(size_t)c * SROWS + sA];
  const int mB = modal[(size_t)c * SROWS + sB];
  const float* qrow0 = &qs[r * LDSTRIDE];
  const float* qrow1 = &qs[(16 + r) * LDSTRIDE];
  const float* drow0 = &dsm[(32 * w + r) * LDSTRIDE];
  const float* drow1 = &dsm[(32 * w + 16 + r) * LDSTRIDE];

  for (int bi = 0; bi < B_PER_WG; ++bi) {
    const int b = b0 + bi;

    // ---- stage q[b] (32x128) asynchronously; reset segmented-max table ----
    const float4* qg = (const float4*)(qemb + (size_t)b * QROWS * DDIM);
    for (int i = tid; i < QROWS * DDIM / 4; i += 256) {
      int row = i >> 5;
      int col = (i & 31) << 2;
      asyncCopy16(&qs[row * LDSTRIDE + col], qg + i);
    }
    if (tid < QROWS * MMOD) mx[tid] = 0u;  // below every encoded finite float
    waitAsync0();
    __syncthreads();

    // ---- WMMA fp32: wave w owns s-tiles {32w, 32w+16}, both q-tiles ----
    // A lane layout (16x4 f32): row = l%16, K = 2*(l>>4) + {0,1}
    // B lane layout (4x16 f32): col = l%16 -> doc row, same K indexing
    v8f acc00 = {}; v8f acc01 = {}; v8f acc10 = {}; v8f acc11 = {};
    for (int ks = 0; ks < DDIM / 4; ++ks) {
      const int kb = 4 * ks + 2 * h;
      v2f a0 = *(const v2f*)(qrow0 + kb);
      v2f a1 = *(const v2f*)(qrow1 + kb);
      v2f bb0 = *(const v2f*)(drow0 + kb);
      v2f bb1 = *(const v2f*)(drow1 + kb);
      acc00 = __builtin_amdgcn_wmma_f32_16x16x4_f32(false, a0, false, bb0, (short)0, acc00, false, false);
      acc01 = __builtin_amdgcn_wmma_f32_16x16x4_f32(false, a0, false, bb1, (short)0, acc01, false, false);
      acc10 = __builtin_amdgcn_wmma_f32_16x16x4_f32(false, a1, false, bb0, (short)0, acc10, false, false);
      acc11 = __builtin_amdgcn_wmma_f32_16x16x4_f32(false, a1, false, bb1, (short)0, acc11, false, false);
    }

    // ---- fused segmented max: C/D element (q = v + 8h [+16], s = tile + r) ----
#pragma unroll
    for (int v = 0; v < 8; ++v) {
      const int qlo = v + 8 * h;
      atomicMax(&mx[qlo * MMOD + mA],        encOrd(acc00[v]));
      atomicMax(&mx[qlo * MMOD + mB],        encOrd(acc01[v]));
      atomicMax(&mx[(16 + qlo) * MMOD + mA], encOrd(acc10[v]));
      atomicMax(&mx[(16 + qlo) * MMOD + mB], encOrd(acc11[v]));
    }
    __syncthreads();

    // ---- per-q decode: agg max (m=1..4, every modality present), masks ----
    if (tid < QROWS) {
      const int q = tid;
      float pm[MMOD];
#pragma unroll
      for (int m = 0; m < MMOD; ++m) pm[m] = decOrd(mx[q * MMOD + m]);
      const float agg = fmaxf(fmaxf(pm[1], pm[2]), fmaxf(pm[3], pm[4]));
      const float mk  = qmask[(size_t)b * QROWS + q] ? 1.0f : 0.0f;
      red[q] = agg * mk;
#pragma unroll
      for (int m = 0; m < MMOD; ++m) red[QROWS + m * QROWS + q] = pm[m];
      red[QROWS + MMOD * QROWS + q] = mk;
    }
    __syncthreads();

    if (tid == 0) {
      float qn = 0.f;
      for (int q = 0; q < QROWS; ++q) qn += red[QROWS + MMOD * QROWS + q];
      const float inv = 1.0f / fmaxf(qn, 1.0f);
      float s = 0.f;
      for (int q = 0; q < QROWS; ++q) s += red[q];
      simg[(size_t)b * BATCH + c] = s * inv;
      for (int m = 0; m < MMOD; ++m) {
        float sm = 0.f;
        for (int q = 0; q < QROWS; ++q) sm += red[QROWS + m * QROWS + q];
        simsg[((size_t)b * BATCH + c) * MMOD + m] = sm * inv;
      }
    }
    // No extra barrier needed: qs was last read before the post-atomic
    // barrier; mx was last read before the post-decode barrier; red is only
    // touched by wave 0 (in-order with its own next-iteration decode).
  }
}

// Per-row logsumexp over {sim_fixed (64)} U {kept (doc,modality) pairs (255)}.
// logsumexp is order-invariant, so excluded columns contribute -inf.
__global__ __launch_bounds__(128) void row_loss_kernel(
    const float* __restrict__ simg, const float* __restrict__ simsg,
    const int* __restrict__ qtypes, float* __restrict__ rowloss) {
  __shared__ float sd[128];
  const int b  = blockIdx.x;
  const int t  = threadIdx.x;
  const int qt = qtypes[b];
  const float posRaw = simsg[((size_t)b * BATCH + b) * MMOD + qt];
  const float pos    = posRaw * INV_TAU;
  const int NCOLS = BATCH + BATCH * MMOD;   // 64 + 320 virtual columns

  float lmax = -3.0e38f;
  for (int col = t; col < NCOLS; col += 128) {
    float v;
    if (col < BATCH) {
      v = ((col == b) ? posRaw : simg[(size_t)b * BATCH + col]) * INV_TAU;
    } else {
      const int j = col - BATCH;
      const int m = j % MMOD, c = j / MMOD;
      v = (m == 0 || j == b * MMOD + qt)
            ? -3.0e38f
            : simsg[((size_t)b * BATCH + c) * MMOD + m] * INV_TAU;
    }
    lmax = fmaxf(lmax, v);
  }
  sd[t] = lmax;
  __syncthreads();
  for (int off = 64; off > 0; off >>= 1) {
    if (t < off) sd[t] = fmaxf(sd[t], sd[t + off]);
    __syncthreads();
  }
  const float bmax = sd[0];
  __syncthreads();

  float ls = 0.f;
  for (int col = t; col < NCOLS; col += 128) {
    float v; bool valid = true;
    if (col < BATCH) {
      v = ((col == b) ? posRaw : simg[(size_t)b * BATCH + col]) * INV_TAU;
    } else {
      const int j = col - BATCH;
      const int m = j % MMOD, c = j / MMOD;
      if (m == 0 || j == b * MMOD + qt) valid = false;
      else v = simsg[((size_t)b * BATCH + c) * MMOD + m] * INV_TAU;
    }
    if (valid) ls += expf(v - bmax);
  }
  sd[t] = ls;
  __syncthreads();
  for (int off = 64; off > 0; off >>= 1) {
    if (t < off) sd[t] += sd[t + off];
    __syncthreads();
  }
  if (t == 0) rowloss[b] = (logf(sd[0]) + bmax - pos) * (1.0f / (float)BATCH);
}

__global__ void final_sum_kernel(const float* __restrict__ rowloss,
                                 float* __restrict__ out) {
  if (threadIdx.x == 0) {
    float s = 0.f;
    for (int i = 0; i < BATCH; ++i) s += rowloss[i];
    out[0] = s;
  }
}

extern "C" void kernel_launch(void* const* d_in, const int* in_sizes, int n_in,
                              void* d_out, int out_size, void* d_ws, size_t ws_size,
                              hipStream_t stream) {
  const float*         qemb   = (const float*)d_in[0];
  const float*         demb   = (const float*)d_in[1];
  const int*           modal  = (const int*)d_in[2];
  const int*           qtypes = (const int*)d_in[3];
  const unsigned char* qmask  = (const unsigned char*)d_in[4];  // numpy bool

  float* ws      = (float*)d_ws;
  float* simg    = ws;                  // 64*64
  float* simsg   = ws + BATCH * BATCH;  // 64*64*5
  float* rowloss = ws + BATCH * BATCH + BATCH * BATCH * MMOD;  // 64

  const size_t smem =
      (size_t)(QROWS * LDSTRIDE + SROWS * LDSTRIDE) * sizeof(float) +
      (size_t)(QROWS * MMOD) * sizeof(unsigned) +
      (size_t)(QROWS * (MMOD + 2)) * sizeof(float);   // ~153.6 KB (<320 KB WGP)

  dim3 grid(BATCH, BATCH / B_PER_WG);   // x = c (doc), y = group of 4 b's
  score_kernel<<<grid, 256, smem, stream>>>(qemb, demb, modal, qmask, simg, simsg);
  row_loss_kernel<<<BATCH, 128, 0, stream>>>(simg, simsg, qtypes, rowloss);
  final_sum_kernel<<<1, 32, 0, stream>>>(rowloss, (float*)d_out);

  (void)in_sizes; (void)n_in; (void)out_size; (void)ws_size;
}